// VideoEncoderRNN_2568390443481
// MI455X (gfx1250) — compile-verified
//
#include <hip/hip_runtime.h>
#include <math.h>

// Problem constants (from reference): B=64, T=256, F_IN=2048, H=1024
#define B_    64
#define T_    256
#define FIN_  2048
#define H_    1024
#define NBLOCKS_STEP ((B_ / 16) * (H_ / 16))   // 256 persistent blocks

typedef float v2f __attribute__((ext_vector_type(2)));
typedef float v8f __attribute__((ext_vector_type(8)));

// Async-to-LDS builtin takes int4* in explicit address spaces:
//   void(v4i addrspace(1)*, v4i addrspace(3)*, imm offset, imm cpol)
typedef int v4i_vs __attribute__((__vector_size__(16)));
typedef __attribute__((address_space(1))) v4i_vs gbl_v4i;
typedef __attribute__((address_space(3))) v4i_vs lds_v4i;

// V_WMMA_F32_16X16X4_F32 : D(16x16 f32) = A(16x4 f32) x B(4x16 f32) + C
static __device__ __forceinline__ v8f wmma_f32(v2f a, v2f b, v8f c) {
  return __builtin_amdgcn_wmma_f32_16x16x4_f32(false, a, false, b, (short)0, c,
                                               false, false);
}

#if __has_builtin(__builtin_amdgcn_global_load_async_to_lds_b128)
#define HAVE_ASYNC_LDS 1
#else
#define HAVE_ASYNC_LDS 0
#endif

static __device__ __forceinline__ void wait_async_le4() {
#if __has_builtin(__builtin_amdgcn_s_wait_asynccnt)
  __builtin_amdgcn_s_wait_asynccnt(4);
#else
  asm volatile("s_wait_asynccnt 4" ::: "memory");
#endif
}
static __device__ __forceinline__ void wait_async_le0() {
#if __has_builtin(__builtin_amdgcn_s_wait_asynccnt)
  __builtin_amdgcn_s_wait_asynccnt(0);
#else
  asm volatile("s_wait_asynccnt 0" ::: "memory");
#endif
}

#define KC_  32
#define PAD_ 36

#if HAVE_ASYNC_LDS
// Stage one 64x32 A panel + one 64x32 W panel into LDS via the Async DMA path
// (GLOBAL_LOAD_ASYNC_TO_LDS_B128, tracked with ASYNCcnt). 4 ops per thread.
static __device__ __forceinline__ void stage_panels_async(
    const float* __restrict__ A, const float* __restrict__ W,
    float* As, float* Ws, int tid, int m0, int n0, int K, int k0)
{
  #pragma unroll
  for (int i = 0; i < 2; ++i) {
    const int idx = tid + i * 256;     // float4 index 0..511
    const int r   = idx >> 3;          // row 0..63
    const int c   = (idx & 7) << 2;    // col 0,4,..,28
    __builtin_amdgcn_global_load_async_to_lds_b128(
        (gbl_v4i*)&A[(size_t)(m0 + r) * K + (k0 + c)],
        (lds_v4i*)&As[r * PAD_ + c], 0, 0);
    __builtin_amdgcn_global_load_async_to_lds_b128(
        (gbl_v4i*)&W[(size_t)(n0 + r) * K + (k0 + c)],
        (lds_v4i*)&Ws[r * PAD_ + c], 0, 0);
  }
}
#endif

// ---------------------------------------------------------------------------
// Generic GEMM with transposed weight:  C[M,N] = A[M,K] * W[N,K]^T + bias[N]
// Block: 256 threads (8 waves). Tile: 64(M) x 64(N), K chunk = 32.
// Waves laid out 4(M) x 2(N); each wave owns a 16x32 sub-tile (2 accumulators).
// LDS panels padded to 36 floats/row (bank-conflict-free, 16B aligned rows).
// Double-buffered async staging: iteration i issues chunk i+1's DMA, then
// waits ASYNCcnt<=4 (async loads complete in order, so chunk i's 4 older ops
// are done) and overlaps the in-flight DMA with the WMMA compute on chunk i.
// ---------------------------------------------------------------------------
__global__ __launch_bounds__(256) void gemm_bias_kernel(
    const float* __restrict__ A, const float* __restrict__ W,
    const float* __restrict__ bias, float* __restrict__ C,
    int N, int K)
{
  __shared__ float As[2][64 * PAD_];
  __shared__ float Ws[2][64 * PAD_];

  const int tid  = threadIdx.x;
  const int lane = tid & 31;
  const int wave = tid >> 5;
  const int m0   = blockIdx.x * 64;
  const int n0   = blockIdx.y * 64;
  const int wm   = (wave & 3) * 16;   // wave's M offset inside tile
  const int wn   = (wave >> 2) * 32;  // wave's N offset inside tile
  const int row  = lane & 15;         // A row / B col handled by this lane
  const int kk2  = (lane >> 4) << 1;  // lanes 0-15 -> K{0,1}, 16-31 -> K{2,3}

  v8f acc0 = {};
  v8f acc1 = {};

  const int nchunks = K / KC_;

#if HAVE_ASYNC_LDS
  stage_panels_async(A, W, As[0], Ws[0], tid, m0, n0, K, 0);  // prologue
#endif

  for (int ch = 0; ch < nchunks; ++ch) {
    const int cur = ch & 1;
#if HAVE_ASYNC_LDS
    if (ch + 1 < nchunks) {
      // Issue next chunk's DMA into the other buffer (its readers finished
      // before the previous iteration's end-of-iteration barrier).
      stage_panels_async(A, W, As[cur ^ 1], Ws[cur ^ 1], tid, m0, n0, K,
                         (ch + 1) * KC_);
      wait_async_le4();   // current chunk's 4 older async ops are complete
    } else {
      wait_async_le0();
    }
    __syncthreads();
#else
    #pragma unroll
    for (int i = 0; i < 2; ++i) {
      const int idx = tid + i * 256;
      const int r   = idx >> 3;
      const int c   = (idx & 7) << 2;
      *(float4*)&As[cur][r * PAD_ + c] =
          *(const float4*)&A[(size_t)(m0 + r) * K + (ch * KC_ + c)];
      *(float4*)&Ws[cur][r * PAD_ + c] =
          *(const float4*)&W[(size_t)(n0 + r) * K + (ch * KC_ + c)];
    }
    __syncthreads();
#endif

    #pragma unroll
    for (int ks = 0; ks < KC_; ks += 4) {
      v2f a  = *(const v2f*)&As[cur][(wm + row)      * PAD_ + ks + kk2];
      v2f b0 = *(const v2f*)&Ws[cur][(wn + row)      * PAD_ + ks + kk2];
      v2f b1 = *(const v2f*)&Ws[cur][(wn + 16 + row) * PAD_ + ks + kk2];
      acc0 = wmma_f32(a, b0, acc0);
      acc1 = wmma_f32(a, b1, acc1);
    }
    __syncthreads();  // all waves done reading buffer `cur` before it is refilled
  }

  // C/D layout: VGPR j holds M=j (lanes 0-15) and M=j+8 (lanes 16-31); N=lane%16.
  const int mbase = m0 + wm + ((lane >> 4) << 3);
  const int nc0   = n0 + wn + (lane & 15);
  const int nc1   = nc0 + 16;
  const float bi0 = bias[nc0];
  const float bi1 = bias[nc1];
  #pragma unroll
  for (int j = 0; j < 8; ++j) {
    const int m = mbase + j;
    C[(size_t)m * N + nc0] = acc0[j] + bi0;
    C[(size_t)m * N + nc1] = acc1[j] + bi1;
  }
}

// ---------------------------------------------------------------------------
// Persistent GRU recurrence: one launch runs all T steps.
// Grid: exactly (B/16) x (H/16) = 256 blocks, 1 wave each (all co-resident).
// Device-wide split barrier per step: release fence -> atomicAdd on a
// monotonically increasing counter -> spin (s_sleep) -> acquire fence.
// w_hh (12MB) and h (256KB) stay resident in the 192MB L2 across steps.
// Each gate's K=1024 WMMA chain is split into two accumulators for ILP.
// ---------------------------------------------------------------------------
__global__ __launch_bounds__(32) void gru_persistent_kernel(
    const float* __restrict__ gx,    // [B,T,3H] precomputed input gates
    const float* __restrict__ w_hh,  // [3H,H]
    const float* __restrict__ b_hh,  // [3H]
    const int*   __restrict__ lens,  // [B]
    float* __restrict__ hA,          // [B,H] ping
    float* __restrict__ hB,          // [B,H] pong
    float* __restrict__ out,         // [B,T,H] (d_out)
    float* __restrict__ hidden,      // [B,H]   (d_out tail)
    unsigned* __restrict__ counter)  // grid barrier counter (pre-zeroed)
{
  const int lane = threadIdx.x & 31;
  const int b0   = blockIdx.x * 16;  // batch tile
  const int n0   = blockIdx.y * 16;  // hidden-column tile
  const int row  = lane & 15;
  const int kk2  = (lane >> 4) << 1;

  const float* __restrict__ wr = w_hh + (size_t)(n0 + row) * H_;
  const float* __restrict__ wz = w_hh + (size_t)(H_ + n0 + row) * H_;
  const float* __restrict__ wv = w_hh + (size_t)(2 * H_ + n0 + row) * H_;

  const int   n     = n0 + (lane & 15);
  const int   mbase = b0 + ((lane >> 4) << 3);
  const float bhr   = b_hh[n];
  const float bhz   = b_hh[H_ + n];
  const float bhn   = b_hh[2 * H_ + n];

  float* hin  = hA;
  float* hout = hB;

  for (int t = 0; t < T_; ++t) {
    const float* __restrict__ ha = hin + (size_t)(b0 + row) * H_;

    v8f accR0 = {}, accR1 = {};
    v8f accZ0 = {}, accZ1 = {};
    v8f accN0 = {}, accN1 = {};
    #pragma unroll 2
    for (int k = 0; k < H_; k += 8) {
      v2f a0  = *(const v2f*)&ha[k + kk2];
      v2f a1  = *(const v2f*)&ha[k + 4 + kk2];
      v2f br0 = *(const v2f*)&wr[k + kk2];
      v2f br1 = *(const v2f*)&wr[k + 4 + kk2];
      v2f bz0 = *(const v2f*)&wz[k + kk2];
      v2f bz1 = *(const v2f*)&wz[k + 4 + kk2];
      v2f bn0 = *(const v2f*)&wv[k + kk2];
      v2f bn1 = *(const v2f*)&wv[k + 4 + kk2];
      accR0 = wmma_f32(a0, br0, accR0);
      accZ0 = wmma_f32(a0, bz0, accZ0);
      accN0 = wmma_f32(a0, bn0, accN0);
      accR1 = wmma_f32(a1, br1, accR1);
      accZ1 = wmma_f32(a1, bz1, accZ1);
      accN1 = wmma_f32(a1, bn1, accN1);
    }
    v8f accR = accR0 + accR1;
    v8f accZ = accZ0 + accZ1;
    v8f accN = accN0 + accN1;

    #pragma unroll
    for (int j = 0; j < 8; ++j) {
      const int b = mbase + j;
      const size_t gbase = ((size_t)b * T_ + t) * (3 * H_);
      const float gir = gx[gbase + n];
      const float giz = gx[gbase + H_ + n];
      const float gin = gx[gbase + 2 * H_ + n];

      const float ghr = accR[j] + bhr;
      const float ghz = accZ[j] + bhz;
      const float ghn = accN[j] + bhn;

      const float r  = 1.0f / (1.0f + __expf(-(gir + ghr)));
      const float z  = 1.0f / (1.0f + __expf(-(giz + ghz)));
      const float ng = tanhf(gin + r * ghn);

      const float hp = hin[(size_t)b * H_ + n];
      const float hn = (1.0f - z) * ng + z * hp;

      const bool  m    = t < lens[b];
      const float hsel = m ? hn : hp;
      const float y    = m ? hn : 0.0f;

      hout[(size_t)b * H_ + n]           = hsel;
      out[((size_t)b * T_ + t) * H_ + n] = y;
      if (t == T_ - 1) hidden[(size_t)b * H_ + n] = y;  // == output[:,-1,:]
    }

    // ---- device-wide step barrier ----
    __builtin_amdgcn_fence(__ATOMIC_RELEASE, "agent");   // flush h_out stores
    const unsigned target = (unsigned)NBLOCKS_STEP * (unsigned)(t + 1);
    if (lane == 0) {
      atomicAdd(counter, 1u);
      while (__hip_atomic_load(counter, __ATOMIC_RELAXED,
                               __HIP_MEMORY_SCOPE_AGENT) < target) {
        __builtin_amdgcn_s_sleep(2);
      }
    }
    __builtin_amdgcn_fence(__ATOMIC_ACQUIRE, "agent");   // invalidate for next h reads

    float* tmp = hin; hin = hout; hout = tmp;
  }
}

__global__ void zero_kernel(float* __restrict__ p, int n) {
  const int i = blockIdx.x * blockDim.x + threadIdx.x;
  if (i < n) p[i] = 0.0f;
}

// ---------------------------------------------------------------------------
extern "C" void kernel_launch(void* const* d_in, const int* in_sizes, int n_in,
                              void* d_out, int out_size, void* d_ws,
                              size_t ws_size, hipStream_t stream) {
  (void)in_sizes; (void)n_in; (void)out_size; (void)ws_size;

  const float* input = (const float*)d_in[0];  // [B,T,F_IN]
  const int*   lens  = (const int*)  d_in[1];  // [B]
  const float* W_p   = (const float*)d_in[2];  // [H,F_IN]
  const float* b_p   = (const float*)d_in[3];  // [H]
  const float* w_ih  = (const float*)d_in[4];  // [3H,H]
  const float* w_hh  = (const float*)d_in[5];  // [3H,H]
  const float* b_ih  = (const float*)d_in[6];  // [3H]
  const float* b_hh  = (const float*)d_in[7];  // [3H]

  float* out    = (float*)d_out;                       // [B,T,H]
  float* hidden = out + (size_t)B_ * T_ * H_;          // [B,H]

  // Workspace layout
  float* emb = (float*)d_ws;                           // [B*T, H]      64 MB
  float* gx  = emb + (size_t)B_ * T_ * H_;             // [B*T, 3H]    192 MB
  float* hA  = gx + (size_t)B_ * T_ * 3 * H_;          // [B, H]
  float* hB  = hA + (size_t)B_ * H_;                   // [B, H]
  unsigned* counter = (unsigned*)(hB + (size_t)B_ * H_);

  const int M = B_ * T_;  // 16384

  // emb = input @ W_p^T + b_p   (M=16384, N=1024, K=2048)
  gemm_bias_kernel<<<dim3(M / 64, H_ / 64), 256, 0, stream>>>(
      input, W_p, b_p, emb, H_, FIN_);

  // gx = emb @ w_ih^T + b_ih    (M=16384, N=3072, K=1024)
  gemm_bias_kernel<<<dim3(M / 64, (3 * H_) / 64), 256, 0, stream>>>(
      emb, w_ih, b_ih, gx, 3 * H_, H_);

  // Zero hA (h0), hB, and the grid-barrier counter (2*B*H + 1 elements).
  zero_kernel<<<(2 * B_ * H_ + 1 + 255) / 256, 256, 0, stream>>>(
      hA, 2 * B_ * H_ + 1);

  // One persistent launch runs all 256 recurrence steps.
  gru_persistent_kernel<<<dim3(B_ / 16, H_ / 16), 32, 0, stream>>>(
      gx, w_hh, b_hh, lens, hA, hB, out, hidden, counter);
}